// SkewedLossFunction_Ordinary_40544491274688
// MI455X (gfx1250) — compile-verified
//
#include <hip/hip_runtime.h>

// Skewed-MAE mean over 33.5M f32 pairs: HBM-bound streaming reduction.
// Memory floor @23.3TB/s ~= 11.5us; compute (~7 flop + 1 v_exp_f32 per elem)
// is negligible. Strategy: b128 non-temporal loads (stream > 192MB L2, single
// use), wave32 shuffle reduce, deterministic 2-kernel finish (no f32 atomics).

typedef float f4  __attribute__((ext_vector_type(4)));
typedef float v2f __attribute__((ext_vector_type(2)));
typedef float v8f __attribute__((ext_vector_type(8)));

#define NBLOCKS  2048
#define NTHREADS 256
#define LOG2E    1.44269504088896340736f

// skewed = |p-t| * exp(sign(t-p) * (2t-1)); lam branches are algebraically equal.
__device__ __forceinline__ float skew_term(float p, float t) {
    float d   = p - t;                       // sign(t-p) = -sign(d)
    float ad  = __builtin_fabsf(d);
    float e   = __builtin_fmaf(2.0f, t, -1.0f) * LOG2E;  // lam * log2(e)
    float arg = (d <= 0.0f) ? e : -e;        // d==0 -> ad==0, exp value moot
    return ad * __builtin_amdgcn_exp2f(arg); // v_exp_f32
}

__global__ void skewed_mae_partial(const float* __restrict__ yp,
                                   const float* __restrict__ yt,
                                   float* __restrict__ partials,
                                   int n4, int n) {
    const f4* p4 = (const f4*)yp;
    const f4* t4 = (const f4*)yt;
    int tid    = blockIdx.x * blockDim.x + threadIdx.x;
    int stride = gridDim.x * blockDim.x;

    float acc = 0.0f;
    int i = tid;
    // 2-deep unroll: 4 independent global_load_b128 (TH=NT) in flight per trip.
    for (; i + stride < n4; i += 2 * stride) {
        f4 p0 = __builtin_nontemporal_load(p4 + i);
        f4 t0 = __builtin_nontemporal_load(t4 + i);
        f4 p1 = __builtin_nontemporal_load(p4 + i + stride);
        f4 t1 = __builtin_nontemporal_load(t4 + i + stride);
        acc += skew_term(p0.x, t0.x) + skew_term(p0.y, t0.y)
             + skew_term(p0.z, t0.z) + skew_term(p0.w, t0.w);
        acc += skew_term(p1.x, t1.x) + skew_term(p1.y, t1.y)
             + skew_term(p1.z, t1.z) + skew_term(p1.w, t1.w);
    }
    for (; i < n4; i += stride) {
        f4 p0 = __builtin_nontemporal_load(p4 + i);
        f4 t0 = __builtin_nontemporal_load(t4 + i);
        acc += skew_term(p0.x, t0.x) + skew_term(p0.y, t0.y)
             + skew_term(p0.z, t0.z) + skew_term(p0.w, t0.w);
    }
    // Scalar tail (n % 4), folded into one thread's accumulator.
    if (blockIdx.x == 0 && threadIdx.x == 0) {
        for (int j = n4 * 4; j < n; ++j) acc += skew_term(yp[j], yt[j]);
    }

    // wave32 reduce (warpSize == 32 on gfx1250).
    #pragma unroll
    for (int off = 16; off > 0; off >>= 1) acc += __shfl_xor(acc, off, 32);

    __shared__ float wsum[NTHREADS / 32];
    int wave = threadIdx.x >> 5;
    int lane = threadIdx.x & 31;
    if (lane == 0) wsum[wave] = acc;
    __syncthreads();
    if (wave == 0) {
        float b = (lane < NTHREADS / 32) ? wsum[lane] : 0.0f;
        #pragma unroll
        for (int off = 4; off > 0; off >>= 1) b += __shfl_xor(b, off, 32);
        if (lane == 0) partials[blockIdx.x] = b;
    }
}

// One-wave finalize: sum NBLOCKS partials deterministically, scale by 1/n.
// Cross-lane sum done with V_WMMA_F32_16X16X4_F32 against a ones B-matrix:
// A[m][k] layout (f32 16x4): lanes 0-15 VGPR0 = A[m][0], lanes 16-31 VGPR0 =
// A[m][2]; put lane value in VGPR0, zero in VGPR1 -> D[m][*] = v[m]+v[m+16].
// Lane L(0-15) then holds D rows M=0..7 in c[0..7], lane L+16 holds M=8..15.
__global__ void skewed_mae_finalize(const float* __restrict__ partials,
                                    float* __restrict__ out,
                                    int nP, float inv_n) {
    int lane = threadIdx.x;  // exactly 32 threads, full wave (EXEC all 1s)
    float s = 0.0f;
    for (int i = lane; i < nP; i += 32) s += partials[i];  // uniform trip count

#if defined(__gfx1250__) && __has_builtin(__builtin_amdgcn_wmma_f32_16x16x4_f32)
    v2f a; a[0] = s;    a[1] = 0.0f;
    v2f b; b[0] = 1.0f; b[1] = 1.0f;
    v8f c = {};
    c = __builtin_amdgcn_wmma_f32_16x16x4_f32(
        /*neg_a=*/false, a, /*neg_b=*/false, b,
        /*c_mod=*/(short)0, c, /*reuse_a=*/false, /*reuse_b=*/false);
    float t = c[0] + c[1] + c[2] + c[3] + c[4] + c[5] + c[6] + c[7];
    t += __shfl_xor(t, 16, 32);  // lanes 0-15: rows 0-7; lanes 16-31: rows 8-15
#else
    float t = s;
    #pragma unroll
    for (int off = 16; off > 0; off >>= 1) t += __shfl_xor(t, off, 32);
#endif
    if (lane == 0) out[0] = t * inv_n;
}

extern "C" void kernel_launch(void* const* d_in, const int* in_sizes, int n_in,
                              void* d_out, int out_size, void* d_ws, size_t ws_size,
                              hipStream_t stream) {
    const float* yp = (const float*)d_in[0];
    const float* yt = (const float*)d_in[1];
    float* partials = (float*)d_ws;    // NBLOCKS * 4 bytes = 8 KB
    float* out      = (float*)d_out;
    int n  = in_sizes[0];
    int n4 = n / 4;

    skewed_mae_partial<<<NBLOCKS, NTHREADS, 0, stream>>>(yp, yt, partials, n4, n);
    // n = 2^25 -> 1/n is an exact power of two in f32.
    skewed_mae_finalize<<<1, 32, 0, stream>>>(partials, out, NBLOCKS, 1.0f / (float)n);
}